// GlobalNormFlipFlopCatMod_52578989637852
// MI455X (gfx1250) — compile-verified
//
#include <hip/hip_runtime.h>
#include <math.h>

// ---------------------------------------------------------------------------
// GlobalNormFlipFlopCatMod for MI455X (gfx1250, wave32, WMMA)
//
// K1: fp32 WMMA GEMM (V_WMMA_F32_16X16X4_F32), 2x3 tile register blocking
//     per wave (6 WMMAs per 5 b64 loads) + tanh/exp epilogue
// K2: linear-domain forward scan over T (lane-per-sequence, 8 wave32s)
// K3: logZ subtraction + grouped log-softmax expansion (45 -> 48 channels)
// ---------------------------------------------------------------------------

typedef float v2f __attribute__((ext_vector_type(2)));
typedef float v8f __attribute__((ext_vector_type(8)));

#define DD 256      // reduction dim
#define SC 45       // score columns
#define OC 48       // output channels
#define NTR 40      // transition channels

// ---------------------------------------------------------------------------
// Kernel 1: y = x * W^T + b.
// Block = 128 threads = 4 waves; wave owns 32 rows x 48 cols = 2x3 WMMA tiles.
// A-frag (16x4 f32): lane l -> row = l%16, K-pair = 2*(l/16)      (float2)
// B-frag (4x16 f32): lane l -> col = l%16, K-pair = 2*(l/16)      (float2)
// C/D   (16x16 f32): lane l -> col = l%16, rows v + 8*(l/16), v=0..7
// ---------------------------------------------------------------------------
__global__ __launch_bounds__(128) void gnff_gemm_act(
    const float* __restrict__ x, const float* __restrict__ W,
    const float* __restrict__ b, float* __restrict__ out,
    float* __restrict__ ep, int Mtot) {
  const int lane = threadIdx.x & 31;
  const int wave = threadIdx.x >> 5;        // 0..3
  const int j    = lane & 15;
  const int half = lane >> 4;

  const int base = blockIdx.x * 128 + wave * 32;   // wave's first row

  // A row pointers for the two row tiles (clamped for tail-safe loads)
  int ar0 = base + j;       if (ar0 >= Mtot) ar0 = Mtot - 1;
  int ar1 = base + 16 + j;  if (ar1 >= Mtot) ar1 = Mtot - 1;
  const float* ap0 = x + (size_t)ar0 * DD + half * 2;
  const float* ap1 = x + (size_t)ar1 * DD + half * 2;

  // B row pointers for the three column tiles (clamp cols >= 45)
  const int s0 = j, s1 = 16 + j;
  const int s2c = (32 + j < SC) ? (32 + j) : (SC - 1);
  const float* bp0 = W + (size_t)s0  * DD + half * 2;
  const float* bp1 = W + (size_t)s1  * DD + half * 2;
  const float* bp2 = W + (size_t)s2c * DD + half * 2;

  v8f acc[2][3];
#pragma unroll
  for (int mi = 0; mi < 2; ++mi)
#pragma unroll
    for (int t = 0; t < 3; ++t) acc[mi][t] = (v8f){};

#pragma unroll 4
  for (int kk = 0; kk < DD; kk += 4) {
    const v2f a0 = *(const v2f*)(ap0 + kk);
    const v2f a1 = *(const v2f*)(ap1 + kk);
    const v2f b0 = *(const v2f*)(bp0 + kk);
    const v2f b1 = *(const v2f*)(bp1 + kk);
    const v2f b2 = *(const v2f*)(bp2 + kk);
    acc[0][0] = __builtin_amdgcn_wmma_f32_16x16x4_f32(
        false, a0, false, b0, (short)0, acc[0][0], false, false);
    acc[0][1] = __builtin_amdgcn_wmma_f32_16x16x4_f32(
        false, a0, false, b1, (short)0, acc[0][1], false, false);
    acc[0][2] = __builtin_amdgcn_wmma_f32_16x16x4_f32(
        false, a0, false, b2, (short)0, acc[0][2], false, false);
    acc[1][0] = __builtin_amdgcn_wmma_f32_16x16x4_f32(
        false, a1, false, b0, (short)0, acc[1][0], false, false);
    acc[1][1] = __builtin_amdgcn_wmma_f32_16x16x4_f32(
        false, a1, false, b1, (short)0, acc[1][1], false, false);
    acc[1][2] = __builtin_amdgcn_wmma_f32_16x16x4_f32(
        false, a1, false, b2, (short)0, acc[1][2], false, false);
  }

  // Epilogue: y = acc + b[s]; trans cols -> 5*tanh (+ exp table), mod -> raw
#pragma unroll
  for (int t = 0; t < 3; ++t) {
    const int s = 16 * t + j;
    if (s < SC) {
      const float bv = b[s];
#pragma unroll
      for (int mi = 0; mi < 2; ++mi) {
#pragma unroll
        for (int v = 0; v < 8; ++v) {
          const int m = base + mi * 16 + v + 8 * half;
          if (m < Mtot) {
            const float y = acc[mi][t][v] + bv;
            if (s < NTR) {
              const float tv = 5.0f * tanhf(y);
              out[(size_t)m * OC + s] = tv;                 // trans (pre-logZ)
              if (ep) ep[(size_t)m * NTR + s] = __expf(tv); // exp table
            } else {
              out[(size_t)m * OC + s] = y;                  // raw mod score
            }
          }
        }
      }
    }
  }
}

// ---------------------------------------------------------------------------
// Kernel 2: forward scan. One lane per sequence n; F[j]=exp(fwd[j]) stays
// normalized (sum 1) so everything runs in the linear domain:
//   n1[i] = sum_j F[j]*E[i*8+j]  (i<4) ;  n2[j] = F[j]*E[32+j]+F[4+j]*E[36+j]
//   S = sum(n1)+sum(n2) ; logZ += log(S) ; F' = n/S
// ---------------------------------------------------------------------------
template <bool USEE>
__global__ __launch_bounds__(32) void gnff_scan(
    const float* __restrict__ ep, const float* __restrict__ outw,
    float* __restrict__ logZ, int T, int N) {
  const int n = blockIdx.x * 32 + threadIdx.x;
  if (n >= N) return;

  float F[8];
#pragma unroll
  for (int i = 0; i < 8; ++i) F[i] = 0.125f;
  float lz = 2.0794415416798357f;  // log(8)

  for (int t = 0; t < T; ++t) {
    float e[NTR];
    if (USEE) {
      const float4* p = reinterpret_cast<const float4*>(
          ep + ((size_t)t * N + n) * NTR);
#pragma unroll
      for (int q = 0; q < 10; ++q) {
        float4 v = p[q];
        e[q * 4 + 0] = v.x; e[q * 4 + 1] = v.y;
        e[q * 4 + 2] = v.z; e[q * 4 + 3] = v.w;
      }
    } else {
      const float4* p = reinterpret_cast<const float4*>(
          outw + ((size_t)t * N + n) * OC);
#pragma unroll
      for (int q = 0; q < 10; ++q) {
        float4 v = p[q];
        e[q * 4 + 0] = __expf(v.x); e[q * 4 + 1] = __expf(v.y);
        e[q * 4 + 2] = __expf(v.z); e[q * 4 + 3] = __expf(v.w);
      }
    }

    float n1[4];
#pragma unroll
    for (int i = 0; i < 4; ++i) {
      float acc = F[0] * e[i * 8];
#pragma unroll
      for (int jj = 1; jj < 8; ++jj) acc = fmaf(F[jj], e[i * 8 + jj], acc);
      n1[i] = acc;
    }
    float n2[4];
#pragma unroll
    for (int jj = 0; jj < 4; ++jj)
      n2[jj] = fmaf(F[jj], e[32 + jj], F[4 + jj] * e[36 + jj]);

    const float S = ((n1[0] + n1[1]) + (n1[2] + n1[3])) +
                    ((n2[0] + n2[1]) + (n2[2] + n2[3]));
    lz += logf(S);
    const float inv = 1.0f / S;
#pragma unroll
    for (int i = 0; i < 4; ++i) { F[i] = n1[i] * inv; F[4 + i] = n2[i] * inv; }
  }
  logZ[n] = lz;
}

// ---------------------------------------------------------------------------
// Kernel 3: out[m, 0:40] -= logZ[n]/T ; expand mod cols 40..44 into grouped
// log-softmax channels 40..47 (groups {0,1,2},{0,3},{0},{0,4}).
// ---------------------------------------------------------------------------
__global__ __launch_bounds__(256) void gnff_finalize(
    float* __restrict__ out, const float* __restrict__ logZ,
    int Mtot, int N, float invT) {
  const int m = blockIdx.x * 256 + threadIdx.x;
  if (m >= Mtot) return;
  const float zs = logZ[m % N] * invT;
  float* row = out + (size_t)m * OC;

  float y[SC];
  const float4* p = reinterpret_cast<const float4*>(row);
#pragma unroll
  for (int q = 0; q < 11; ++q) {
    float4 v = p[q];
    y[q * 4 + 0] = v.x; y[q * 4 + 1] = v.y;
    y[q * 4 + 2] = v.z; y[q * 4 + 3] = v.w;
  }
  y[44] = row[44];

#pragma unroll
  for (int cidx = 0; cidx < NTR; ++cidx) row[cidx] = y[cidx] - zs;

  const float a0 = y[40], a1 = y[41], a2 = y[42], a3 = y[43], a4 = y[44];
  const float mx3 = fmaxf(a0, fmaxf(a1, a2));
  const float l3  = mx3 + logf(__expf(a0 - mx3) + __expf(a1 - mx3) +
                               __expf(a2 - mx3));
  const float mxa = fmaxf(a0, a3);
  const float l2a = mxa + logf(__expf(a0 - mxa) + __expf(a3 - mxa));
  const float mxb = fmaxf(a0, a4);
  const float l2b = mxb + logf(__expf(a0 - mxb) + __expf(a4 - mxb));

  row[40] = a0 - l3;  row[41] = a1 - l3;  row[42] = a2 - l3;
  row[43] = a0 - l2a; row[44] = a3 - l2a;
  row[45] = 0.0f;
  row[46] = a0 - l2b; row[47] = a4 - l2b;
}

// ---------------------------------------------------------------------------
extern "C" void kernel_launch(void* const* d_in, const int* in_sizes, int n_in,
                              void* d_out, int out_size, void* d_ws,
                              size_t ws_size, hipStream_t stream) {
  const float* x = (const float*)d_in[0];
  const float* W = (const float*)d_in[1];
  const float* b = (const float*)d_in[2];
  float* out = (float*)d_out;

  const int Mtot = in_sizes[0] / DD;   // T*N = 512000
  const int N    = 256;
  const int T    = Mtot / N;           // 2000

  // workspace: [0,4096) logZ (N floats), rest: exp(trans) table if it fits
  float* logZ = (float*)d_ws;
  const size_t ebytes = (size_t)Mtot * NTR * sizeof(float);
  float* ep = (ws_size >= 4096 + ebytes)
                  ? (float*)((char*)d_ws + 4096) : nullptr;

  dim3 g1((Mtot + 127) / 128), b1(128);
  gnff_gemm_act<<<g1, b1, 0, stream>>>(x, W, b, out, ep, Mtot);

  dim3 g2((N + 31) / 32), b2(32);
  if (ep) gnff_scan<true ><<<g2, b2, 0, stream>>>(ep, out, logZ, T, N);
  else    gnff_scan<false><<<g2, b2, 0, stream>>>(ep, out, logZ, T, N);

  dim3 g3((Mtot + 255) / 256), b3(256);
  gnff_finalize<<<g3, b3, 0, stream>>>(out, logZ, Mtot, N, 1.0f / (float)T);
}